// ExternalMemory_9869834846732
// MI455X (gfx1250) — compile-verified
//
#include <hip/hip_runtime.h>
#include <stdint.h>

typedef float v2f __attribute__((ext_vector_type(2)));
typedef float v4f __attribute__((ext_vector_type(4)));
typedef float v8f __attribute__((ext_vector_type(8)));

#define D_DIM 256
#define TOPK_CAP 8192

// ---------------- init workspace state ----------------
__global__ void init_ws_kernel(uint32_t* state, uint32_t* hist, int kinit) {
  int t = threadIdx.x;
  hist[t] = 0u;
  if (t == 0) {
    state[0] = 0u;                 // radix prefix
    state[1] = 0u;                 // radix mask
    state[2] = (uint32_t)kinit;    // remaining rank k
    state[3] = 0u;                 // candidate count
  }
}

// ---------------- feature reduction, stage 1 (deterministic) ----------------
__global__ void feat_reduce1(const float* __restrict__ feat,
                             float* __restrict__ psum,
                             float* __restrict__ psq, int Brows) {
  __shared__ float red[256];
  int d = threadIdx.x;
  int b = blockIdx.x;
  float s = 0.f, q = 0.f;
  for (int r = 0; r < 8; ++r) {
    int row = b * 8 + r;
    if (row < Brows) {
      float v = feat[(size_t)row * D_DIM + d];
      s += v;
      q += v * v;
    }
  }
  psum[(size_t)b * D_DIM + d] = s;
  red[d] = q;
  __syncthreads();
  for (int st = 128; st > 0; st >>= 1) {
    if (d < st) red[d] += red[d + st];
    __syncthreads();
  }
  if (d == 0) psq[b] = red[0];
}

// ---------------- feature reduction, stage 2 ----------------
__global__ void feat_reduce2(const float* __restrict__ psum,
                             const float* __restrict__ psq,
                             float* __restrict__ fsum,
                             float* __restrict__ fsq, int nblk) {
  __shared__ float red[256];
  int d = threadIdx.x;
  float s = 0.f;
  for (int b = 0; b < nblk; ++b) s += psum[(size_t)b * D_DIM + d];
  fsum[d] = s;
  float q = 0.f;
  for (int b = d; b < nblk; b += 256) q += psq[b];
  red[d] = q;
  __syncthreads();
  for (int st = 128; st > 0; st >>= 1) {
    if (d < st) red[d] += red[d + st];
    __syncthreads();
  }
  if (d == 0) fsq[0] = red[0];
}

// ---------------- WMMA scoring kernel ----------------
// One wave per 16-row tile of `memory`.  Each macro-iteration covers 8
// k-values with ONE global_load_b128 per lane (half 0: k0..k0+3, half 1:
// k0+4..k0+7) feeding TWO v_wmma_f32_16x16x4_f32 ops (contraction over K is
// symmetric, so A and B just need a consistent k-slot assignment):
//   wmma0 contracts {k0,k0+1,k0+4,k0+5}, wmma1 contracts {k0+2,k0+3,k0+6,k0+7}.
// B columns are all identical (= f_sum chunk, staged in LDS), so every lane's
// accumulator column is cross[row]. VALU FMAs accumulate sum-of-squares
// concurrently with the matrix pipe.
__global__ void score_wmma_kernel(const float* __restrict__ mem,
                                  const float* __restrict__ fsum,
                                  const float* __restrict__ fsq,
                                  float* __restrict__ scores,
                                  int Mrows, float Bf) {
  __shared__ __align__(16) float fsum_lds[D_DIM];
  __shared__ float cross_lds[8 * 16];

  fsum_lds[threadIdx.x] = fsum[threadIdx.x];   // blockDim.x == 256
  __syncthreads();

  int lane = threadIdx.x & 31;
  int wave = threadIdx.x >> 5;
  int tile = blockIdx.x * (blockDim.x >> 5) + wave;
  int r16  = lane & 15;
  int half = lane >> 4;

  long row   = (long)tile * 16 + r16;
  bool rowok = row < (long)Mrows;
  long rowc  = rowok ? row : (long)(Mrows - 1);   // clamp: keep loads unconditional

  v8f acc0 = {};
  v8f acc1 = {};
  float msq = 0.f;

  if ((long)tile * 16 < (long)Mrows) {   // wave-uniform activity guard
    const float* rp = mem + (size_t)rowc * D_DIM + half * 4;
    const float* bp = fsum_lds + half * 4;
#pragma unroll 4
    for (int k0 = 0; k0 < D_DIM; k0 += 8) {
      v4f a = *(const v4f*)(rp + k0);    // global_load_b128 (unconditional)
      v4f b = *(const v4f*)(bp + k0);    // ds_load_b128
      v2f a01; a01.x = a.x; a01.y = a.y;
      v2f a23; a23.x = a.z; a23.y = a.w;
      v2f b01; b01.x = b.x; b01.y = b.y;
      v2f b23; b23.x = b.z; b23.y = b.w;
      acc0 = __builtin_amdgcn_wmma_f32_16x16x4_f32(
          false, a01, false, b01, (short)0, acc0, false, false);
      acc1 = __builtin_amdgcn_wmma_f32_16x16x4_f32(
          false, a23, false, b23, (short)0, acc1, false, false);
      msq += a.x * a.x + a.y * a.y + a.z * a.z + a.w * a.w;
    }
    v8f acc = acc0 + acc1;
    msq += __shfl_xor(msq, 16, 32);   // merge K-halves: lanes 0-15 own rows 0-15

    float* cl = cross_lds + wave * 16;
    if (lane == 0) {   // lane 0 holds D[M=0..7][N=0] in acc[0..7]
      cl[0] = acc[0]; cl[1] = acc[1]; cl[2] = acc[2]; cl[3] = acc[3];
      cl[4] = acc[4]; cl[5] = acc[5]; cl[6] = acc[6]; cl[7] = acc[7];
    }
    if (lane == 16) {  // lane 16 holds D[M=8..15][N=0] in acc[0..7]
      cl[8]  = acc[0]; cl[9]  = acc[1]; cl[10] = acc[2]; cl[11] = acc[3];
      cl[12] = acc[4]; cl[13] = acc[5]; cl[14] = acc[6]; cl[15] = acc[7];
    }
  }
  __syncthreads();
  if (lane < 16 && rowok && (long)tile * 16 < (long)Mrows) {
    float cr = cross_lds[wave * 16 + lane];
    scores[row] = -(Bf * msq - 2.f * cr + fsq[0]);
  }
}

// ---------------- radix-select helpers ----------------
__device__ __forceinline__ uint32_t score_key(float s) {
  uint32_t u = __float_as_uint(s);
  return (u & 0x80000000u) ? ~u : (u | 0x80000000u);  // monotonic: bigger f -> bigger key
}

__global__ void hist_kernel(const float* __restrict__ scores,
                            const uint32_t* __restrict__ state,
                            uint32_t* __restrict__ hist, int Mrows, int shift) {
  __shared__ uint32_t lh[256];
  lh[threadIdx.x] = 0u;
  __syncthreads();
  uint32_t prefix = state[0], mask = state[1];
  for (int i = blockIdx.x * blockDim.x + threadIdx.x; i < Mrows;
       i += gridDim.x * blockDim.x) {
    uint32_t key = score_key(scores[i]);
    if ((key & mask) == prefix)
      atomicAdd(&lh[(key >> shift) & 0xFFu], 1u);
  }
  __syncthreads();
  atomicAdd(&hist[threadIdx.x], lh[threadIdx.x]);
}

__global__ void scan_kernel(uint32_t* state, uint32_t* hist, int shift) {
  uint32_t kk = state[2];
  uint32_t digit = 0u;
  for (int b = 255; b >= 0; --b) {
    uint32_t c = hist[b];
    if (kk <= c) { digit = (uint32_t)b; break; }
    kk -= c;
  }
  state[0] |= digit << shift;
  state[1] |= 0xFFu << shift;
  state[2] = kk;
  for (int b = 0; b < 256; ++b) hist[b] = 0u;
}

// ---------------- collect candidates >= threshold ----------------
__global__ void collect_kernel(const float* __restrict__ scores,
                               uint32_t* __restrict__ state,
                               uint32_t* __restrict__ ckey,
                               uint32_t* __restrict__ cidx, int Mrows) {
  uint32_t thresh = state[0];
  for (int i = blockIdx.x * blockDim.x + threadIdx.x; i < Mrows;
       i += gridDim.x * blockDim.x) {
    uint32_t key = score_key(scores[i]);
    if (key >= thresh) {
      uint32_t pos = atomicAdd(&state[3], 1u);
      if (pos < TOPK_CAP) { ckey[pos] = key; cidx[pos] = (uint32_t)i; }
    }
  }
}

// ---------------- exact rank + gather (order-independent, deterministic) ----
__global__ void rank_gather_kernel(const float* __restrict__ mem,
                                   const uint32_t* __restrict__ ckey,
                                   const uint32_t* __restrict__ cidx,
                                   const uint32_t* __restrict__ state,
                                   float* __restrict__ out, int K) {
  __shared__ uint32_t red[256];
  uint32_t C = state[3];
  if (C > TOPK_CAP) C = TOPK_CAP;
  uint32_t b = blockIdx.x;
  if (b >= C) return;
  uint32_t kb = ckey[b], ib = cidx[b];
  uint32_t t = threadIdx.x;
  uint32_t r = 0u;
  for (uint32_t j = t; j < C; j += 256u) {
    uint32_t kj = ckey[j], ij = cidx[j];
    r += (kj > kb || (kj == kb && ij < ib)) ? 1u : 0u;
  }
  red[t] = r;
  __syncthreads();
  for (int st = 128; st > 0; st >>= 1) {
    if (t < (uint32_t)st) red[t] += red[t + st];
    __syncthreads();
  }
  uint32_t rank = red[0];
  if (rank < (uint32_t)K) {
    out[(size_t)rank * D_DIM + t] = mem[(size_t)ib * D_DIM + t];
  }
}

// ---------------- host launch ----------------
extern "C" void kernel_launch(void* const* d_in, const int* in_sizes, int n_in,
                              void* d_out, int out_size, void* d_ws, size_t ws_size,
                              hipStream_t stream) {
  (void)n_in; (void)out_size; (void)ws_size;
  const float* feat = (const float*)d_in[0];
  const float* mem  = (const float*)d_in[1];
  float* out = (float*)d_out;

  int Brows = in_sizes[0] / D_DIM;                 // 1024
  int Mrows = in_sizes[1] / D_DIM;                 // 100000
  int K = (Brows < Mrows) ? Brows : Mrows;         // 1024

  uint32_t* ws    = (uint32_t*)d_ws;
  uint32_t* state = ws;                            // 4
  uint32_t* hist  = ws + 4;                        // 256
  float* fsum     = (float*)(ws + 260);            // 256
  float* fsq      = (float*)(ws + 516);            // 1
  float* psq      = (float*)(ws + 520);            // up to 512
  float* psum     = (float*)(ws + 1032);           // FB * 256
  int FB = (Brows + 7) / 8;
  float* scores   = (float*)(ws + 1032) + (size_t)FB * D_DIM;
  uint32_t* ckey  = (uint32_t*)(scores + Mrows);
  uint32_t* cidx  = ckey + TOPK_CAP;

  init_ws_kernel<<<1, 256, 0, stream>>>(state, hist, K);
  feat_reduce1<<<FB, 256, 0, stream>>>(feat, psum, psq, Brows);
  feat_reduce2<<<1, 256, 0, stream>>>(psum, psq, fsum, fsq, FB);

  int tiles  = (Mrows + 15) / 16;
  int blocks = (tiles + 7) / 8;                    // 8 waves per 256-thread block
  score_wmma_kernel<<<blocks, 256, 0, stream>>>(mem, fsum, fsq, scores, Mrows,
                                                (float)Brows);

  for (int pass = 0; pass < 4; ++pass) {
    int shift = 24 - 8 * pass;
    hist_kernel<<<256, 256, 0, stream>>>(scores, state, hist, Mrows, shift);
    scan_kernel<<<1, 1, 0, stream>>>(state, hist, shift);
  }

  collect_kernel<<<256, 256, 0, stream>>>(scores, state, ckey, cidx, Mrows);
  rank_gather_kernel<<<TOPK_CAP, 256, 0, stream>>>(mem, ckey, cidx, state, out, K);
}